// RRWPLinearEdgeEncoder_69690139345167
// MI455X (gfx1250) — compile-verified
//
#include <hip/hip_runtime.h>

typedef __attribute__((ext_vector_type(2))) float v2f;
typedef __attribute__((ext_vector_type(8))) float v8f;

#define NGRAPH 16
#define NPG    256
#define EMB    16
#define OUTD   64
#define MTOT   (NGRAPH * NPG * NPG)   /* 1,048,576 dense rows */
#define ESP    65536                  /* sparse edges */

// ---------------------------------------------------------------------------
// Kernel 1: dense[i,:] = rrwp[i,:] @ W.T via V_WMMA_F32_16X16X4_F32, plus
// emit out_idx (cast to float). One wave32 owns a 16-row stripe x 64 cols.
// Grid: 8192 blocks x 256 thr = 65536 waves = MTOT/16; also exactly 2*MTOT
// threads, one per out_idx element.
// ---------------------------------------------------------------------------
__global__ __launch_bounds__(256) void rrwp_gemm_idx_kernel(
    const float* __restrict__ rrwp,   // [MTOT, EMB] row-major
    const float* __restrict__ W,      // [OUTD, EMB] row-major
    float* __restrict__ out)          // [2*MTOT idx | MTOT*OUTD dense]
{
    const unsigned tid  = threadIdx.x;
    const unsigned lane = tid & 31u;
    const unsigned wave = blockIdx.x * 8u + (tid >> 5);
    const unsigned l15  = lane & 15u;
    const unsigned koff = (lane >> 4) << 1;   // K offset: 0 (lanes 0-15) / 2 (16-31)

    // ---- out_idx as float: t in [0, 2*MTOT) exactly covers both rows ----
    {
        const unsigned t = blockIdx.x * 256u + tid;
        float v;
        if (t < (unsigned)MTOT) {
            v = (float)(t >> 8);                          // src = i / N
        } else {
            const unsigned u = t - (unsigned)MTOT;
            v = (float)(((u >> 16) << 8) | (u & 255u));   // dst = (i/N^2)*N + i%N
        }
        out[t] = v;
    }

    float* __restrict__ dense = out + (size_t)2 * MTOT;
    const unsigned rowBase = wave * 16u;

    // ---- A slices: a[kk] holds rrwp[rowBase+l15][kk*4+koff .. +1] (16x4 f32) ----
    v2f a[4];
    const float* arow = rrwp + (size_t)(rowBase + l15) * EMB + koff;
#pragma unroll
    for (int kk = 0; kk < 4; ++kk)
        a[kk] = *(const v2f*)(arow + kk * 4);

    const unsigned mhalf = (lane >> 4) << 3;   // rows 0-7 vs 8-15 of the C tile

    // ---- 4 N-tiles of 16 columns; K=16 = 4 chained 16x16x4 WMMAs each ----
#pragma unroll
    for (int nt = 0; nt < 4; ++nt) {
        const float* wrow = W + (size_t)(nt * 16 + l15) * EMB + koff;
        v8f c = {0.f, 0.f, 0.f, 0.f, 0.f, 0.f, 0.f, 0.f};
#pragma unroll
        for (int kk = 0; kk < 4; ++kk) {
            const v2f b = *(const v2f*)(wrow + kk * 4);   // B = W^T tile (4x16)
            c = __builtin_amdgcn_wmma_f32_16x16x4_f32(
                    /*neg_a=*/false, a[kk], /*neg_b=*/false, b,
                    /*c_mod=*/(short)0, c, /*reuse_a=*/false, /*reuse_b=*/false);
        }
        // store 16x16 tile: per VGPR v, lanes 0-15 write one 64B row chunk
#pragma unroll
        for (int v = 0; v < 8; ++v) {
            const unsigned m = (unsigned)v + mhalf;
            dense[(size_t)(rowBase + m) * OUTD + (unsigned)(nt * 16) + l15] = c[v];
        }
    }
}

// ---------------------------------------------------------------------------
// Kernel 2: scatter-add edge_attr onto dense rows (collisions -> f32 atomics).
// flat = src*256 + (dst & 255). One thread per (edge, out-dim).
// ---------------------------------------------------------------------------
__global__ __launch_bounds__(256) void rrwp_edge_scatter_kernel(
    const int* __restrict__ edge_index,   // [2, ESP] (row0 = src, row1 = dst)
    const float* __restrict__ edge_attr,  // [ESP, OUTD]
    float* __restrict__ out)
{
    const unsigned idx = blockIdx.x * 256u + threadIdx.x;  // ESP*OUTD threads
    const unsigned e = idx >> 6;
    const unsigned o = idx & 63u;
    const int src = edge_index[e];
    const int dst = edge_index[ESP + e];
    const unsigned flat = (unsigned)src * 256u + ((unsigned)dst & 255u);
    float* dense = out + (size_t)2 * MTOT;
    atomicAdd(dense + (size_t)flat * OUTD + o, edge_attr[(size_t)e * OUTD + o]);
}

extern "C" void kernel_launch(void* const* d_in, const int* in_sizes, int n_in,
                              void* d_out, int out_size, void* d_ws, size_t ws_size,
                              hipStream_t stream) {
    // setup_inputs order: rrwp_index, rrwp_val, edge_index, edge_attr, batch, W, ...
    const float* rrwp  = (const float*)d_in[1];
    const int*   eidx  = (const int*)d_in[2];
    const float* eattr = (const float*)d_in[3];
    const float* W     = (const float*)d_in[5];
    float* out = (float*)d_out;

    // 65536 waves (MTOT/16), 8 waves/block
    rrwp_gemm_idx_kernel<<<MTOT / 16 / 8, 256, 0, stream>>>(rrwp, W, out);
    // 65536 edges x 64 dims
    rrwp_edge_scatter_kernel<<<(ESP * OUTD) / 256, 256, 0, stream>>>(eidx, eattr, out);
}